// SentenceDifference_25881472926220
// MI455X (gfx1250) — compile-verified
//
#include <hip/hip_runtime.h>

typedef float v2f __attribute__((ext_vector_type(2)));
typedef float v8f __attribute__((ext_vector_type(8)));

#define LDIM  128          // left tokens
#define RDIM  128          // right tokens
#define BDIM  256          // batch
#define DDIM  768          // feature dim
#define CHUNK 32           // K-chunk staged in LDS
#define LSTR  36           // LDS row stride (floats) for staged chunks (16B aligned, conflict-free)
#define ASTR  129          // LDS row stride (floats) for attention (conflict-free column reads)

// LDS layout (floats):
//   [0 .. 128*ASTR)                : attention (written AFTER staging is dead; staging aliased below)
//   [0 .. 128*LSTR)                : ldsL chunk   (alias, phase-1 only)
//   [128*LSTR .. 2*128*LSTR)      : ldsR chunk   (alias, phase-1 only)
//   [128*ASTR .. +128)            : idxR (int)
//   [128*ASTR+128 .. +128)        : idxL (int)
#define SMEM_FLOATS (LDIM * ASTR)
#define SMEM_BYTES  (SMEM_FLOATS * 4 + 2 * LDIM * 4)

__global__ __launch_bounds__(256) void sentdiff_kernel(
    const float* __restrict__ left,  const int* __restrict__ llenp,
    const float* __restrict__ right, const int* __restrict__ rlenp,
    float* __restrict__ outLD, float* __restrict__ outRD,
    float* __restrict__ outLM, float* __restrict__ outRM)
{
    const int b    = blockIdx.x;
    const int tid  = threadIdx.x;
    const int wave = tid >> 5;        // 0..7 : owns attention rows 16*wave..16*wave+15
    const int lane = tid & 31;
    const int half = lane >> 4;       // 0/1 : K-pair selector per WMMA f32 A/B layout
    const int lr   = lane & 15;       // M (A) / N (B) index within tile

    extern __shared__ float smem[];
    float* attn = smem;                    // 128 x 129
    float* ldsL = smem;                    // aliased: 128 x 36  (phase 1 only)
    float* ldsR = smem + LDIM * LSTR;      // aliased: 128 x 36  (phase 1 only)
    int*   idxR = (int*)(smem + SMEM_FLOATS);
    int*   idxL = idxR + LDIM;

    const int Llen = llenp[b];
    const int Rlen = rlenp[b];

    const size_t rowstride = (size_t)BDIM * DDIM;        // stride between tokens l -> l+1
    const float* lbase = left  + (size_t)b * DDIM;       // left[l,b,d]  = lbase + l*rowstride + d
    const float* rbase = right + (size_t)b * DDIM;

    // ---------------- Phase 1: attention = left x right^T via V_WMMA_F32_16X16X4_F32 -------------
    v8f acc[8] = {};   // 8 column tiles of 16x16 f32 for this wave's row strip

    for (int kc = 0; kc < DDIM / CHUNK; ++kc) {
        __syncthreads();
        // Stage 128x32 chunks of left and right into LDS (coalesced float4, 4 iters, no divergence)
        for (int i = tid; i < LDIM * (CHUNK / 4); i += 256) {
            const int row = i >> 3;
            const int q   = (i & 7) * 4;
            const float4 lv = *(const float4*)(lbase + (size_t)row * rowstride + kc * CHUNK + q);
            *(float4*)(ldsL + row * LSTR + q) = lv;
            const float4 rv = *(const float4*)(rbase + (size_t)row * rowstride + kc * CHUNK + q);
            *(float4*)(ldsR + row * LSTR + q) = rv;
        }
        __syncthreads();

        #pragma unroll
        for (int kk = 0; kk < CHUNK; kk += 4) {
            const int kloc = kk + 2 * half;   // lanes 0-15 -> K=k,k+1 ; lanes 16-31 -> K=k+2,k+3
            const v2f a = *(const v2f*)(ldsL + (16 * wave + lr) * LSTR + kloc);
            #pragma unroll
            for (int nt = 0; nt < 8; ++nt) {
                const v2f bf = *(const v2f*)(ldsR + (16 * nt + lr) * LSTR + kloc);
                acc[nt] = __builtin_amdgcn_wmma_f32_16x16x4_f32(
                    /*neg_a=*/false, a, /*neg_b=*/false, bf,
                    /*c_mod=*/(short)0, acc[nt], /*reuse_a=*/false, /*reuse_b=*/false);
            }
        }
    }
    __syncthreads();   // staging dead; safe to write attention over the aliased region

    // C/D layout: lanes 0-15: VGPR v -> (M=v, N=lane); lanes 16-31: VGPR v -> (M=v+8, N=lane-16)
    #pragma unroll
    for (int nt = 0; nt < 8; ++nt) {
        const int   col = 16 * nt + lr;
        const float rm  = (col < Rlen) ? 1.0f : 0.0f;
        #pragma unroll
        for (int v = 0; v < 8; ++v) {
            const int   row = 16 * wave + v + 8 * half;
            const float lm  = (row < Llen) ? 1.0f : 0.0f;
            attn[row * ASTR + col] = acc[nt][v] * lm * rm;
        }
    }
    __syncthreads();

    // ---------------- Phase 2: argmax (first-occurrence, matching jnp.argmax) + masks -----------
    if (tid < LDIM) {
        const int l = tid;
        float best = attn[l * ASTR];
        int   bi   = 0;
        for (int r = 1; r < RDIM; ++r) {
            const float v = attn[l * ASTR + r];
            if (v > best) { best = v; bi = r; }
        }
        idxR[l] = bi;
        outLM[b * LDIM + l] = (l < Llen) ? 1.0f : 0.0f;
    } else {
        const int r = tid - LDIM;
        float best = attn[r];
        int   bi   = 0;
        for (int l = 1; l < LDIM; ++l) {
            const float v = attn[l * ASTR + r];
            if (v > best) { best = v; bi = l; }
        }
        idxL[r] = bi;
        outRM[b * RDIM + r] = (r < Rlen) ? 1.0f : 0.0f;
    }
    __syncthreads();

    // ---------------- Phase 3: masked differences (coalesced float4) -----------------------------
    // left_difference[l,b,:] = (left[l,b,:] - right[idxR[l],b,:]) * lmask(l)
    for (int i = tid; i < LDIM * (DDIM / 4); i += 256) {
        const int   l  = i / (DDIM / 4);
        const int   q  = (i % (DDIM / 4)) * 4;
        const float lm = (l < Llen) ? 1.0f : 0.0f;
        const int   ir = idxR[l];
        const float4 lv = *(const float4*)(lbase + (size_t)l  * rowstride + q);
        const float4 rv = *(const float4*)(rbase + (size_t)ir * rowstride + q);
        float4 o;
        o.x = (lv.x - rv.x) * lm;  o.y = (lv.y - rv.y) * lm;
        o.z = (lv.z - rv.z) * lm;  o.w = (lv.w - rv.w) * lm;
        *(float4*)(outLD + ((size_t)l * BDIM + b) * DDIM + q) = o;
    }
    // right_difference[r,b,:] = (right[r,b,:] - left[idxL[r],b,:]) * rmask(r)
    for (int i = tid; i < RDIM * (DDIM / 4); i += 256) {
        const int   r  = i / (DDIM / 4);
        const int   q  = (i % (DDIM / 4)) * 4;
        const float rm = (r < Rlen) ? 1.0f : 0.0f;
        const int   il = idxL[r];
        const float4 rv = *(const float4*)(rbase + (size_t)r  * rowstride + q);
        const float4 lv = *(const float4*)(lbase + (size_t)il * rowstride + q);
        float4 o;
        o.x = (rv.x - lv.x) * rm;  o.y = (rv.y - lv.y) * rm;
        o.z = (rv.z - lv.z) * rm;  o.w = (rv.w - lv.w) * rm;
        *(float4*)(outRD + ((size_t)r * BDIM + b) * DDIM + q) = o;
    }
}

extern "C" void kernel_launch(void* const* d_in, const int* in_sizes, int n_in,
                              void* d_out, int out_size, void* d_ws, size_t ws_size,
                              hipStream_t stream) {
    (void)in_sizes; (void)n_in; (void)out_size; (void)d_ws; (void)ws_size;
    const float* left  = (const float*)d_in[0];   // (L,B,D) fp32
    const int*   llen  = (const int*)  d_in[1];   // (B,)
    const float* right = (const float*)d_in[2];   // (R,B,D) fp32
    const int*   rlen  = (const int*)  d_in[3];   // (B,)

    float* out   = (float*)d_out;
    float* outLD = out;                                            // (L,B,D)
    float* outRD = outLD + (size_t)LDIM * BDIM * DDIM;             // (R,B,D)
    float* outLM = outRD + (size_t)RDIM * BDIM * DDIM;             // (B,L)
    float* outRM = outLM + (size_t)BDIM * LDIM;                    // (B,R)

    sentdiff_kernel<<<BDIM, 256, SMEM_BYTES, stream>>>(
        left, llen, right, rlen, outLD, outRD, outLM, outRM);
}